// FiberBundleConvFixedGraphMLP_39513699123537
// MI455X (gfx1250) — compile-verified
//
#include <hip/hip_runtime.h>
#include <math.h>

#define KS 5
#define NVEC 12
#define CINC 8
#define COUTC 8
#define NCH 96            // CIN*NV
#define TAPS 125
#define VOX (64*64*64)

typedef __attribute__((ext_vector_type(2))) float v2f;
typedef __attribute__((ext_vector_type(8))) float v8f;

__device__ __forceinline__ float gelu_exact(float v) {
    return 0.5f * v * (1.0f + erff(v * 0.7071067811865476f));
}

// ---------------------------------------------------------------------------
// Kernel 1a: spatial (depthwise) weights.  sw[(ci*12+v)*125 + t]
// ---------------------------------------------------------------------------
__global__ void gen_spatial_weights(const float* __restrict__ sp,
                                    const float* __restrict__ w1, const float* __restrict__ b1,
                                    const float* __restrict__ w2, const float* __restrict__ b2,
                                    const float* __restrict__ w3,
                                    float* __restrict__ sw) {
    int e = blockIdx.x * blockDim.x + threadIdx.x;
    if (e >= TAPS * NVEC) return;
    float a[2] = { sp[2 * e], sp[2 * e + 1] };
    float feat[14];
    feat[0] = a[0]; feat[1] = a[1];
    int idx = 2;
    for (int p = 0; p < 2; ++p)
        for (int q = 0; q < 2; ++q) feat[idx++] = a[p] * a[q];
    for (int p = 0; p < 2; ++p)
        for (int q = 0; q < 2; ++q)
            for (int r = 0; r < 2; ++r) feat[idx++] = a[p] * a[q] * a[r];
    float h1[8];
    for (int o = 0; o < 8; ++o) {
        float s = b1[o];
        for (int i = 0; i < 14; ++i) s = fmaf(w1[o * 14 + i], feat[i], s);
        h1[o] = gelu_exact(s);
    }
    float h2[8];
    for (int o = 0; o < 8; ++o) {
        float s = b2[o];
        for (int i = 0; i < 8; ++i) s = fmaf(w2[o * 8 + i], h1[i], s);
        h2[o] = gelu_exact(s);
    }
    int t = e / NVEC, v = e % NVEC;
    for (int ci = 0; ci < CINC; ++ci) {
        float s = 0.f;
        for (int i = 0; i < 8; ++i) s = fmaf(w3[ci * 8 + i], h2[i], s);
        sw[(ci * NVEC + v) * TAPS + t] = s;
    }
}

// ---------------------------------------------------------------------------
// Kernel 1b: fiber weights W[i=g*12+w][j=f*12+v] = mlp(sph)[v*12+w][f*8+g],
// stored directly as WMMA f32 16x16x4 A-fragments: afrag[m][kk][lane][slot].
//   lane<16:  M=lane,   slot0=K(4kk+0) slot1=K(4kk+1)
//   lane>=16: M=lane-16, slot0=K(4kk+2) slot1=K(4kk+3)
// ---------------------------------------------------------------------------
__global__ void gen_fiber_weights(const float* __restrict__ sph,
                                  const float* __restrict__ w1, const float* __restrict__ b1,
                                  const float* __restrict__ w2, const float* __restrict__ b2,
                                  const float* __restrict__ w3,
                                  float* __restrict__ afrag) {
    int r = blockIdx.x * blockDim.x + threadIdx.x;
    if (r >= NVEC * NVEC) return;
    float s = sph[r];
    float feat[3] = { s, s * s, s * s * s };
    float h1[8];
    for (int o = 0; o < 8; ++o) {
        float t = b1[o];
        for (int i = 0; i < 3; ++i) t = fmaf(w1[o * 3 + i], feat[i], t);
        h1[o] = gelu_exact(t);
    }
    float h2[8];
    for (int o = 0; o < 8; ++o) {
        float t = b2[o];
        for (int i = 0; i < 8; ++i) t = fmaf(w2[o * 8 + i], h1[i], t);
        h2[o] = gelu_exact(t);
    }
    int v = r / NVEC, w = r % NVEC;
    for (int f = 0; f < CINC; ++f) {
        for (int g = 0; g < COUTC; ++g) {
            float val = 0.f;
            for (int i = 0; i < 8; ++i) val = fmaf(w3[(f * COUTC + g) * 8 + i], h2[i], val);
            int ii = g * NVEC + w;          // output row
            int jj = f * NVEC + v;          // contraction col
            int m = ii >> 4, row = ii & 15;
            int kk = jj >> 2, ko = jj & 3;
            int lane = (ko < 2) ? row : (16 + row);
            int slot = ko & 1;
            afrag[(((m * 24 + kk) * 32) + lane) * 2 + slot] = val;
        }
    }
}

// ---------------------------------------------------------------------------
// Kernel 2: depthwise 5x5x5 conv (SAME, cross-correlation), y written to d_out.
// grid: ((b*96+c)*64 + x)*16 + ytile ; block 256: 4 y-rows x 64 z.
// ---------------------------------------------------------------------------
__global__ void depthwise_conv(const float* __restrict__ x,
                               const float* __restrict__ sw,
                               float* __restrict__ y) {
    int blk = blockIdx.x;
    int ytile = blk & 15;
    int xcoord = (blk >> 4) & 63;
    int bc = blk >> 10;                    // b*96 + c
    int c = bc % NCH;

    __shared__ float wsm[TAPS];
    if (threadIdx.x < TAPS) wsm[threadIdx.x] = sw[c * TAPS + threadIdx.x];
    __syncthreads();

    int z = threadIdx.x & 63;
    int yy0 = ytile * 4 + (threadIdx.x >> 6);
    const float* xc = x + (size_t)bc * VOX;
    float acc = 0.f;
    for (int dx = 0; dx < 5; ++dx) {
        int xx = xcoord + dx - 2;
        if (xx < 0 || xx >= 64) continue;          // uniform across block
        for (int dy = 0; dy < 5; ++dy) {
            int yy = yy0 + dy - 2;
            bool yok = (yy >= 0) && (yy < 64);
            const float* row = xc + ((size_t)xx * 64 + (yok ? yy : 0)) * 64;
            #pragma unroll
            for (int dz = 0; dz < 5; ++dz) {
                int zz = z + dz - 2;
                float xv = (yok && zz >= 0 && zz < 64) ? row[zz] : 0.f;
                acc = fmaf(wsm[dx * 25 + dy * 5 + dz], xv, acc);
            }
        }
    }
    y[(size_t)bc * VOX + ((size_t)xcoord * 64 + yy0) * 64 + z] = acc;
}

// ---------------------------------------------------------------------------
// Kernel 3: per-voxel 96x96 fiber mix via V_WMMA_F32_16X16X4_F32, in place.
// Block = 256 threads = 8 waves; each wave owns 16 contiguous voxels.
// Wave loads its full 96x16 y column block into registers BEFORE any store,
// so reusing d_out as the y buffer is race-free.
// ---------------------------------------------------------------------------
__global__ void fiber_mix(float* __restrict__ io,
                          const float* __restrict__ afrag,
                          const float* __restrict__ bias) {
    int lane = threadIdx.x & 31;
    int wave = threadIdx.x >> 5;
    int tilesPerBatch = VOX / 128;
    int b = blockIdx.x / tilesPerBatch;
    int p0 = (blockIdx.x % tilesPerBatch) * 128 + wave * 16;

    float* base = io + (size_t)b * NCH * VOX;
    int col = p0 + (lane & 15);
    int jbase = (lane < 16) ? 0 : 2;

    // Stage B fragments: y[j][col] for all 24 K-steps (48 regs / lane).
    float yreg[48];
    #pragma unroll
    for (int kk = 0; kk < 24; ++kk) {
        yreg[2 * kk]     = base[(size_t)(4 * kk + jbase) * VOX + col];
        yreg[2 * kk + 1] = base[(size_t)(4 * kk + jbase + 1) * VOX + col];
    }

    #pragma unroll
    for (int m = 0; m < 6; ++m) {
        v8f c = { 0.f, 0.f, 0.f, 0.f, 0.f, 0.f, 0.f, 0.f };
        #pragma unroll
        for (int kk = 0; kk < 24; ++kk) {
            v2f a = *(const v2f*)(afrag + ((m * 24 + kk) * 32 + lane) * 2);
            v2f bf;
            bf[0] = yreg[2 * kk];
            bf[1] = yreg[2 * kk + 1];
            c = __builtin_amdgcn_wmma_f32_16x16x4_f32(
                    false, a, false, bf, (short)0, c, false, false);
        }
        int rowBase = m * 16 + ((lane < 16) ? 0 : 8);
        #pragma unroll
        for (int r = 0; r < 8; ++r) {
            int i = rowBase + r;
            base[(size_t)i * VOX + col] = c[r] + bias[i / NVEC];
        }
    }
}

// ---------------------------------------------------------------------------
extern "C" void kernel_launch(void* const* d_in, const int* in_sizes, int n_in,
                              void* d_out, int out_size, void* d_ws, size_t ws_size,
                              hipStream_t stream) {
    const float* x         = (const float*)d_in[0];
    const float* spatial   = (const float*)d_in[1];
    const float* spherical = (const float*)d_in[2];
    const float* bw1 = (const float*)d_in[3];
    const float* bb1 = (const float*)d_in[4];
    const float* bw2 = (const float*)d_in[5];
    const float* bb2 = (const float*)d_in[6];
    const float* bw3 = (const float*)d_in[7];
    const float* fw1 = (const float*)d_in[8];
    const float* fb1 = (const float*)d_in[9];
    const float* fw2 = (const float*)d_in[10];
    const float* fb2 = (const float*)d_in[11];
    const float* fw3 = (const float*)d_in[12];
    const float* bias = (const float*)d_in[13];

    float* out   = (float*)d_out;
    float* ws    = (float*)d_ws;
    float* sw    = ws;            // 96*125 = 12000 floats
    float* afrag = ws + 12000;    // 6*24*32*2 = 9216 floats

    gen_spatial_weights<<<(TAPS * NVEC + 255) / 256, 256, 0, stream>>>(
        spatial, bw1, bb1, bw2, bb2, bw3, sw);
    gen_fiber_weights<<<1, 256, 0, stream>>>(
        spherical, fw1, fb1, fw2, fb2, fw3, afrag);

    // y -> d_out (in place intermediate)
    depthwise_conv<<<2 * NCH * 64 * 16, 256, 0, stream>>>(x, sw, out);

    // 96x96 mix per voxel, in place on d_out
    fiber_mix<<<2 * (VOX / 128), 256, 0, stream>>>(out, afrag, bias);
}